// SNDCGLoss_32530082300104
// MI455X (gfx1250) — compile-verified
//
#include <hip/hip_runtime.h>

typedef __attribute__((ext_vector_type(2))) float v2f;
typedef __attribute__((ext_vector_type(8))) float v8f;

#define NN 256
#define DD 1024
#define EE 300
#define KCAP 5
#define TAU_INV 10.0f

// ---------------------------------------------------------------------------
// L2-normalize rows: rows 0..1279 -> v_text_emb [N*K,E]; 1280..1535 -> t_text_emb [N,E]
// One wave per row.
// ---------------------------------------------------------------------------
__global__ void norm_rows_kernel(const float* __restrict__ vte,
                                 const float* __restrict__ tte,
                                 float* __restrict__ ve_n,
                                 float* __restrict__ te_n) {
    int row = blockIdx.x;
    int lane = threadIdx.x;
    const float* src;
    float* dst;
    if (row < NN * KCAP) { src = vte + (size_t)row * EE; dst = ve_n + (size_t)row * EE; }
    else { int r = row - NN * KCAP; src = tte + (size_t)r * EE; dst = te_n + (size_t)r * EE; }
    float ss = 0.f;
    for (int e = lane; e < EE; e += 32) { float x = src[e]; ss += x * x; }
    #pragma unroll
    for (int off = 16; off > 0; off >>= 1) ss += __shfl_xor(ss, off, 32);
    float inv = rsqrtf(ss);
    for (int e = lane; e < EE; e += 32) dst[e] = src[e] * inv;
}

// ---------------------------------------------------------------------------
// scores = v @ t^T via V_WMMA_F32_16X16X4_F32. One wave per 16x16 tile.
// A layout (16x4 f32): lanes 0-15: M=lane, VGPR0/1 = K0/K1; lanes 16-31: K2/K3.
// -> per-lane A frag = float2 at v[(i0+m)*D + k + 2*half].
// B (4x16) mirrored: per-lane frag = float2 at t[(j0+n)*D + k + 2*half].
// C (16x16 f32): vgpr r -> row m = r + 8*half, col n = lane&15.
// ---------------------------------------------------------------------------
__global__ void scores_wmma_kernel(const float* __restrict__ v,
                                   const float* __restrict__ t,
                                   float* __restrict__ scores) {
    int tile = blockIdx.x;
    int i0 = (tile >> 4) << 4;
    int j0 = (tile & 15) << 4;
    int lane = threadIdx.x;
    int half = lane >> 4;
    int l = lane & 15;
    const float* arow = v + (size_t)(i0 + l) * DD + 2 * half;
    const float* brow = t + (size_t)(j0 + l) * DD + 2 * half;
    v8f acc = {};
    #pragma unroll 4
    for (int k = 0; k < DD; k += 4) {
        v2f a = *(const v2f*)(arow + k);
        v2f b = *(const v2f*)(brow + k);
        acc = __builtin_amdgcn_wmma_f32_16x16x4_f32(false, a, false, b,
                                                    (short)0, acc, false, false);
    }
    #pragma unroll
    for (int r = 0; r < 8; ++r)
        scores[(size_t)(i0 + r + 8 * half) * NN + (j0 + l)] = acc[r];
}

// ---------------------------------------------------------------------------
// relevance[i,j] = max over cap of (ve_n[i,cap] . te_n[j]); then (1+r)/2 with
// diag forced to 1; gain = 2^rel - 1. One wave per 16x16 tile, fp32 WMMA.
// ---------------------------------------------------------------------------
__global__ void relevance_wmma_kernel(const float* __restrict__ ve_n,
                                      const float* __restrict__ te_n,
                                      float* __restrict__ rel,
                                      float* __restrict__ gain) {
    int tile = blockIdx.x;
    int i0 = (tile >> 4) << 4;
    int j0 = (tile & 15) << 4;
    int lane = threadIdx.x;
    int half = lane >> 4;
    int l = lane & 15;
    const float* brow = te_n + (size_t)(j0 + l) * EE + 2 * half;
    v8f mx;
    #pragma unroll
    for (int r = 0; r < 8; ++r) mx[r] = -2.0f;
    for (int cap = 0; cap < KCAP; ++cap) {
        const float* arow = ve_n + ((size_t)(i0 + l) * KCAP + cap) * EE + 2 * half;
        v8f acc = {};
        #pragma unroll 5
        for (int k = 0; k < EE; k += 4) {   // 300 = 75 * 4, exact
            v2f a = *(const v2f*)(arow + k);
            v2f b = *(const v2f*)(brow + k);
            acc = __builtin_amdgcn_wmma_f32_16x16x4_f32(false, a, false, b,
                                                        (short)0, acc, false, false);
        }
        #pragma unroll
        for (int r = 0; r < 8; ++r) mx[r] = fmaxf(mx[r], acc[r]);
    }
    #pragma unroll
    for (int r = 0; r < 8; ++r) {
        int i = i0 + r + 8 * half;
        int j = j0 + l;
        float rv = (i == j) ? 1.0f : 0.5f * (1.0f + mx[r]);
        rel[(size_t)i * NN + j]  = rv;
        gain[(size_t)i * NN + j] = exp2f(rv) - 1.0f;
    }
}

// ---------------------------------------------------------------------------
// idcg[i] = sum_j gain[i,j] / log2(1 + hard_rank(relevance[i,:], j))
// One 256-thread block per row; row cached in LDS.
// ---------------------------------------------------------------------------
__global__ void idcg_kernel(const float* __restrict__ rel,
                            const float* __restrict__ gain,
                            float* __restrict__ idcg) {
    __shared__ float srow[NN];
    __shared__ float red[NN];
    int i = blockIdx.x, j = threadIdx.x;
    srow[j] = rel[(size_t)i * NN + j];
    __syncthreads();
    float rj = srow[j];
    int cnt = 0;
    for (int k = 0; k < NN; ++k) cnt += (srow[k] > rj) ? 1 : 0;
    float rk = (float)cnt + 1.0f;
    red[j] = gain[(size_t)i * NN + j] / log2f(1.0f + rk);
    __syncthreads();
    for (int s = NN / 2; s > 0; s >>= 1) {
        if (j < s) red[j] += red[j + s];
        __syncthreads();
    }
    if (j == 0) idcg[i] = red[0];
}

// ---------------------------------------------------------------------------
// dcg / real_dcg for one direction. transposed=0: rows of scores; =1: columns.
// soft rank: sum_{k!=j} sigmoid((s[k]-s[j])/tau) with clip(+-50) on exponent.
// ---------------------------------------------------------------------------
__global__ void dcg_kernel(const float* __restrict__ scores,
                           const float* __restrict__ gain,
                           float* __restrict__ dcg,
                           float* __restrict__ real_dcg,
                           int transposed) {
    __shared__ float srow[NN];
    __shared__ float red[NN];
    __shared__ float red2[NN];
    int i = blockIdx.x, j = threadIdx.x;
    srow[j] = transposed ? scores[(size_t)j * NN + i] : scores[(size_t)i * NN + j];
    __syncthreads();
    float sj = srow[j];
    float soft = 0.f;
    int realc = 0;
    for (int k = 0; k < NN; ++k) {
        float diff = srow[k] - sj;
        realc += (diff > 0.f) ? 1 : 0;
        float e = fminf(fmaxf(-diff * TAU_INV, -50.f), 50.f);
        float sig = 1.0f / (1.0f + __expf(e));
        soft += (k != j) ? sig : 0.f;
    }
    float g = gain[(size_t)i * NN + j];
    red[j]  = g / log2f(2.0f + soft);          // log2(1 + (soft_sum + 1))
    red2[j] = g / log2f(2.0f + (float)realc);  // log2(1 + (cnt + 1))
    __syncthreads();
    for (int s = NN / 2; s > 0; s >>= 1) {
        if (j < s) { red[j] += red[j + s]; red2[j] += red2[j + s]; }
        __syncthreads();
    }
    if (j == 0) { dcg[i] = red[0]; real_dcg[i] = red2[0]; }
}

// ---------------------------------------------------------------------------
// Final: ndcg means -> 5 scalars.
// ---------------------------------------------------------------------------
__global__ void final_kernel(const float* __restrict__ idcg,
                             const float* __restrict__ dcg_t,
                             const float* __restrict__ rdcg_t,
                             const float* __restrict__ dcg_v,
                             const float* __restrict__ rdcg_v,
                             float* __restrict__ out) {
    __shared__ float a[NN], b[NN], c[NN], d[NN];
    int j = threadIdx.x;
    float inv = 1.0f / idcg[j];
    a[j] = dcg_t[j] * inv;
    b[j] = rdcg_t[j] * inv;
    c[j] = dcg_v[j] * inv;
    d[j] = rdcg_v[j] * inv;
    __syncthreads();
    for (int s = NN / 2; s > 0; s >>= 1) {
        if (j < s) { a[j] += a[j+s]; b[j] += b[j+s]; c[j] += c[j+s]; d[j] += d[j+s]; }
        __syncthreads();
    }
    if (j == 0) {
        const float invn = 1.0f / (float)NN;
        float mt = a[0] * invn, rt = b[0] * invn, mv = c[0] * invn, rv = d[0] * invn;
        out[0] = 0.5f * ((1.0f - mt) + (1.0f - mv));
        out[1] = mt;
        out[2] = rt;
        out[3] = mv;
        out[4] = rv;
    }
}

extern "C" void kernel_launch(void* const* d_in, const int* in_sizes, int n_in,
                              void* d_out, int out_size, void* d_ws, size_t ws_size,
                              hipStream_t stream) {
    (void)in_sizes; (void)n_in; (void)out_size; (void)ws_size;
    const float* v   = (const float*)d_in[0];   // [256,1024]
    const float* t   = (const float*)d_in[1];   // [256,1024]
    const float* vte = (const float*)d_in[2];   // [256,5,300]
    const float* tte = (const float*)d_in[3];   // [256,300]
    float* out = (float*)d_out;                 // 5 scalars

    float* ws     = (float*)d_ws;
    float* scores = ws;                          // 65536
    float* rel    = scores + NN * NN;            // 65536
    float* gain   = rel + NN * NN;               // 65536
    float* ve_n   = gain + NN * NN;              // 384000
    float* te_n   = ve_n + NN * KCAP * EE;       // 76800
    float* idcg   = te_n + NN * EE;              // 256
    float* dcg_t  = idcg + NN;
    float* rdcg_t = dcg_t + NN;
    float* dcg_v  = rdcg_t + NN;
    float* rdcg_v = dcg_v + NN;                  // total ~2.52 MB

    norm_rows_kernel<<<NN * KCAP + NN, 32, 0, stream>>>(vte, tte, ve_n, te_n);
    scores_wmma_kernel<<<256, 32, 0, stream>>>(v, t, scores);
    relevance_wmma_kernel<<<256, 32, 0, stream>>>(ve_n, te_n, rel, gain);
    idcg_kernel<<<NN, NN, 0, stream>>>(rel, gain, idcg);
    dcg_kernel<<<NN, NN, 0, stream>>>(scores, gain, dcg_t, rdcg_t, 0);
    dcg_kernel<<<NN, NN, 0, stream>>>(scores, gain, dcg_v, rdcg_v, 1);
    final_kernel<<<1, NN, 0, stream>>>(idcg, dcg_t, rdcg_t, dcg_v, rdcg_v, out);
}